// ChamferLossLayer_33105607917718
// MI455X (gfx1250) — compile-verified
//
#include <hip/hip_runtime.h>

typedef float v2f __attribute__((ext_vector_type(2)));
typedef float v8f __attribute__((ext_vector_type(8)));

#define TILE 16
#define WPB  8   // waves per block (256 threads, wave32)

// Pack a cloud (NB, P, 3) into WMMA B-fragment lane order, one v2f per lane:
//   tile t, lane l      (l<16):  (x[n], y[n])          n = t*16 + l   (K=0,1)
//   tile t, lane 16+l:           (z[n], |b_n|^2)                      (K=2,3)
__global__ __launch_bounds__(256) void pack_bfrag(
    const float* __restrict__ cloud,  // (NB, P, 3)
    v2f* __restrict__ pk,             // (NB, P/16, 32) v2f
    int P, int NB)
{
    int i = blockIdx.x * blockDim.x + threadIdx.x;
    if (i >= NB * P) return;
    int b = i / P;
    int n = i % P;
    int t = n >> 4;
    int l = n & 15;
    float x = cloud[(size_t)i * 3 + 0];
    float y = cloud[(size_t)i * 3 + 1];
    float z = cloud[(size_t)i * 3 + 2];
    v2f* o = pk + (size_t)b * P * 2;   // P/16 tiles * 32 lanes = P*2 v2f
    v2f lo; lo.x = x; lo.y = y;
    v2f hi; hi.x = z; hi.y = x * x + y * y + z * z;
    o[t * 32 + l]      = lo;
    o[t * 32 + 16 + l] = hi;
}

// One wave computes min_j ||a_i - b_j||^2 for 16 consecutive query rows i,
// sweeping all target points. Distance tile from a single
// v_wmma_f32_16x16x4_f32:  A=[-2x,-2y,-2z,1], B=[x,y,z,|b|^2], C=|a|^2.
__global__ __launch_bounds__(256) void chamfer_min_pass(
    const float* __restrict__ Ac,   // (NB, PA, 3) query cloud (raw)
    const v2f*   __restrict__ Bpk,  // (NB, PB/16, 32) packed target fragments
    float* __restrict__ minOut,     // (NB, PA)  min sq-dist per query point
    int PA, int PB)
{
    const int lane  = threadIdx.x & 31;
    const int wave  = threadIdx.x >> 5;
    const int batch = blockIdx.y;
    const int r0    = (blockIdx.x * WPB + wave) * TILE;
    if (r0 >= PA) return;   // wave-uniform; EXEC all-1 within live waves

    const float* __restrict__ A  = Ac  + (size_t)batch * PA * 3;
    const v2f*   __restrict__ Bp = Bpk + (size_t)batch * PB * 2;

    const bool hi = (lane >= 16);
    const int  l  = lane & 15;

    // ---- A fragment (loop invariant): row m = r0 + l
    // lanes 0-15: K=0,1 -> (-2x, -2y);  lanes 16-31: K=2,3 -> (-2z, 1)
    float ax = A[(r0 + l) * 3 + 0];
    float ay = A[(r0 + l) * 3 + 1];
    float az = A[(r0 + l) * 3 + 2];
    v2f afrag;
    afrag.x = hi ? (-2.0f * az) : (-2.0f * ax);
    afrag.y = hi ? 1.0f         : (-2.0f * ay);

    // ---- C fragment (loop invariant): C[m][n] = |a_m|^2,  m = r0 + v + 8*hi
    v8f cfrag;
#pragma unroll
    for (int v = 0; v < 8; ++v) {
        int m = r0 + v + (hi ? 8 : 0);
        float x = A[m * 3 + 0], y = A[m * 3 + 1], z = A[m * 3 + 2];
        cfrag[v] = x * x + y * y + z * z;
    }

    v8f acc;
#pragma unroll
    for (int v = 0; v < 8; ++v) acc[v] = 3.0e38f;

    // Hot loop: 1x global_load_b64 + 1x wmma + 8x v_min_num_f32.
    // Clamp-at-zero is deferred: min_j max(d_j,0) == max(min_j d_j, 0).
    const int ntiles = PB / TILE;
    const v2f* __restrict__ p = Bp + lane;
    for (int t = 0; t < ntiles; ++t) {
        v2f bfrag = p[t * 32];
        v8f d = __builtin_amdgcn_wmma_f32_16x16x4_f32(
            false, afrag, false, bfrag, (short)0, cfrag, false, false);
#pragma unroll
        for (int v = 0; v < 8; ++v)
            acc[v] = fminf(acc[v], d[v]);
    }

    // Reduce over the 16 columns held across lanes of each half-wave,
    // then apply the deferred numerical floor.
#pragma unroll
    for (int v = 0; v < 8; ++v) {
        float r = acc[v];
        r = fminf(r, __shfl_xor(r, 1, 32));
        r = fminf(r, __shfl_xor(r, 2, 32));
        r = fminf(r, __shfl_xor(r, 4, 32));
        r = fminf(r, __shfl_xor(r, 8, 32));
        if (l == 0) {
            int m = r0 + v + (hi ? 8 : 0);
            minOut[(size_t)batch * PA + m] = fmaxf(r, 0.0f);
        }
    }
}

__global__ __launch_bounds__(256) void chamfer_finalize(
    const float* __restrict__ min1, const float* __restrict__ min2,
    float* __restrict__ out, int P1, int P2)
{
    __shared__ float s1[256];
    __shared__ float s2[256];
    const int b = blockIdx.x;
    float a1 = 0.0f, a2 = 0.0f;
    for (int i = threadIdx.x; i < P1; i += blockDim.x) a1 += min1[(size_t)b * P1 + i];
    for (int i = threadIdx.x; i < P2; i += blockDim.x) a2 += min2[(size_t)b * P2 + i];
    s1[threadIdx.x] = a1;
    s2[threadIdx.x] = a2;
    __syncthreads();
    for (int off = 128; off > 0; off >>= 1) {
        if ((int)threadIdx.x < off) {
            s1[threadIdx.x] += s1[threadIdx.x + off];
            s2[threadIdx.x] += s2[threadIdx.x + off];
        }
        __syncthreads();
    }
    if (threadIdx.x == 0)
        out[b] = s1[0] / (float)P1 + s2[0] / (float)P2;
}

extern "C" void kernel_launch(void* const* d_in, const int* in_sizes, int n_in,
                              void* d_out, int out_size, void* d_ws, size_t ws_size,
                              hipStream_t stream) {
    const float* c1 = (const float*)d_in[0];   // (2, P1, 3) fp32
    const float* c2 = (const float*)d_in[1];   // (2, P2, 3) fp32
    float* out = (float*)d_out;                // (2,) fp32

    const int NB = 2, D = 3;
    const int P1 = in_sizes[0] / (NB * D);
    const int P2 = in_sizes[1] / (NB * D);

    // Workspace layout (floats): min1 | min2 | pk1 (P1*4 per batch) | pk2
    float* min1 = (float*)d_ws;                       // NB*P1
    float* min2 = min1 + (size_t)NB * P1;             // NB*P2
    v2f*   pk1  = (v2f*)(min2 + (size_t)NB * P2);     // NB*P1*2 v2f
    v2f*   pk2  = pk1 + (size_t)NB * P1 * 2;          // NB*P2*2 v2f

    dim3 blk(256);

    // Pack both clouds into WMMA B-fragment layout.
    pack_bfrag<<<dim3((NB * P1 + 255) / 256), blk, 0, stream>>>(c1, pk1, P1, NB);
    pack_bfrag<<<dim3((NB * P2 + 255) / 256), blk, 0, stream>>>(c2, pk2, P2, NB);

    dim3 g1((P1 + TILE * WPB - 1) / (TILE * WPB), NB);
    dim3 g2((P2 + TILE * WPB - 1) / (TILE * WPB), NB);

    // Pass 1: for each point of cloud1, min over cloud2.
    chamfer_min_pass<<<g1, blk, 0, stream>>>(c1, pk2, min1, P1, P2);
    // Pass 2 (swapped): for each point of cloud2, min over cloud1.
    chamfer_min_pass<<<g2, blk, 0, stream>>>(c2, pk1, min2, P2, P1);
    // Means per batch, sum of both directions.
    chamfer_finalize<<<dim3(NB), blk, 0, stream>>>(min1, min2, out, P1, P2);
}